// LaneATTHead_90838558310857
// MI455X (gfx1250) — compile-verified
//
#include <hip/hip_runtime.h>

typedef __attribute__((ext_vector_type(2))) float v2f;
typedef __attribute__((ext_vector_type(8))) float v8f;

#define FH 45
#define FW 80
#define HW (FH*FW)          // 3600
#define CIN 64
#define CF  64
#define NB  16
#define NANCH 2784
#define NTILES (NANCH/16)   // 174
#define KDIM (CF*FH)        // 2880
#define JCOLS 75            // 2 cls + 73 reg
#define JPAD 80
#define OUTC 77

// ---------------- Kernel 0: pack [cls_w | reg_w | 0] into (KDIM x JPAD) ----------------
__global__ __launch_bounds__(256) void pack_w_kernel(const float* __restrict__ cls_w,
                                                     const float* __restrict__ reg_w,
                                                     float* __restrict__ wp) {
    int i = blockIdx.x * 256 + threadIdx.x;
    if (i >= KDIM * JPAD) return;
    int k = i / JPAD, j = i % JPAD;
    float v = 0.0f;
    if (j < 2)          v = cls_w[k * 2 + j];
    else if (j < JCOLS) v = reg_w[k * 73 + (j - 2)];
    wp[i] = v;
}

// ---------------- Kernel 1: 1x1 conv as WMMA f32 GEMM: feat = conv_w @ x + b ----------------
// One wave computes a 64(M=out chan) x 16(hw) tile; K = 64.
__global__ __launch_bounds__(256) void conv1x1_kernel(const float* __restrict__ x,
                                                      const float* __restrict__ w,
                                                      const float* __restrict__ bias,
                                                      float* __restrict__ feat) {
    int wid  = blockIdx.x * 8 + (threadIdx.x >> 5);   // 3600 waves total
    int lane = threadIdx.x & 31;
    int b    = wid / 225;
    int t    = wid % 225;
    int hw0  = t * 16;
    int ll   = lane & 15;       // M (A) / N (B) index within tile
    int lh   = lane >> 4;       // K-half selector
    const float* xb = x + b * CIN * HW;

    v8f acc[4] = {};
    for (int k = 0; k < CIN; k += 4) {
        int k0 = k + lh * 2;
        v2f bv;
        bv.x = xb[(k0    ) * HW + hw0 + ll];
        bv.y = xb[(k0 + 1) * HW + hw0 + ll];
#pragma unroll
        for (int m = 0; m < 4; ++m) {
            v2f av;
            av.x = w[(m * 16 + ll) * CIN + k0    ];
            av.y = w[(m * 16 + ll) * CIN + k0 + 1];
            acc[m] = __builtin_amdgcn_wmma_f32_16x16x4_f32(false, av, false, bv,
                                                           (short)0, acc[m], false, false);
        }
    }
#pragma unroll
    for (int m = 0; m < 4; ++m) {
#pragma unroll
        for (int r = 0; r < 8; ++r) {
            int o = m * 16 + r + lh * 8;
            feat[b * CF * HW + o * HW + hw0 + ll] = acc[m][r] + bias[o];
        }
    }
}

// ---------------- Kernel 2: fused gather + GEMM (f @ [cls|reg]) + epilogue ----------------
// One wave per (batch-pair, 16-anchor tile). The two batches share the W (B-matrix)
// fragments and the LDS gather-offset table -> 10 WMMAs per 10 W-loads per k-step.
__global__ __launch_bounds__(256) void head_gemm_kernel(const float* __restrict__ feat,
                                                        const float* __restrict__ wp,
                                                        const float* __restrict__ cls_b,
                                                        const float* __restrict__ reg_b,
                                                        const float* __restrict__ anchors,
                                                        const int* __restrict__ cut_xs,
                                                        const unsigned char* __restrict__ invalid,
                                                        float* __restrict__ out) {
    __shared__ int hofs[8 * 16 * FH];                 // per-wave gather-offset tables
    int wslot = threadIdx.x >> 5;
    int lane  = threadIdx.x & 31;
    int wid   = blockIdx.x * 8 + wslot;               // 1392 waves total
    int bp    = wid / NTILES;                         // batch pair 0..7
    int nt    = wid % NTILES;
    int n0    = nt * 16;
    int b0    = bp * 2;
    int ll    = lane & 15;
    int lh    = lane >> 4;

    // Stage per-anchor offsets: off[n][h] = invalid ? -1 : h*FW + xs[n][h]  (batch-independent)
    int* myofs = hofs + wslot * 16 * FH;
    for (int i = lane; i < 16 * FH; i += 32) {
        int nn  = n0 + i / FH;
        int h   = i % FH;
        int xi  = cut_xs[nn * FH + h];
        myofs[i] = invalid[nn * FH + h] ? -1 : (h * FW + xi);
    }
    __syncthreads();

    const float* fb0  = feat + (b0    ) * CF * HW;
    const float* fb1  = feat + (b0 + 1) * CF * HW;
    const int*   lofs = myofs + ll * FH;              // this lane's anchor row

    v8f acc0[5] = {};                                 // batch b0, 5 j-tiles
    v8f acc1[5] = {};                                 // batch b0+1
    int h = lh * 2;     // k = ks*4 + lh*2 (+v);  c = k / FH, h = k % FH, tracked incrementally
    int c = 0;
    int wbase = lh * 2 * JPAD;
    for (int ks = 0; ks < KDIM / 4; ++ks) {
        int h0 = h,      c0 = c;
        int h1 = h0 + 1, c1 = c0;
        if (h1 >= FH) { h1 -= FH; ++c1; }
        int o0 = lofs[h0];
        int o1 = lofs[h1];
        int e0 = c0 * HW + o0;
        int e1 = c1 * HW + o1;
        v2f av0, av1;
        av0.x = (o0 >= 0) ? fb0[e0] : 0.0f;
        av0.y = (o1 >= 0) ? fb0[e1] : 0.0f;
        av1.x = (o0 >= 0) ? fb1[e0] : 0.0f;
        av1.y = (o1 >= 0) ? fb1[e1] : 0.0f;
#pragma unroll
        for (int jt = 0; jt < 5; ++jt) {
            v2f bv;
            bv.x = wp[wbase +        jt * 16 + ll];
            bv.y = wp[wbase + JPAD + jt * 16 + ll];
            acc0[jt] = __builtin_amdgcn_wmma_f32_16x16x4_f32(false, av0, false, bv,
                                                             (short)0, acc0[jt], false, false);
            acc1[jt] = __builtin_amdgcn_wmma_f32_16x16x4_f32(false, av1, false, bv,
                                                             (short)0, acc1[jt], false, false);
        }
        h += 4; if (h >= FH) { h -= FH; ++c; }
        wbase += 4 * JPAD;
    }

    // Epilogue: C layout -> VGPR r, lane<16: M=r ; lane>=16: M=r+8 ; N = ll
#pragma unroll
    for (int bb = 0; bb < 2; ++bb) {
        float* ob = out + (size_t)(b0 + bb) * NANCH * OUTC;
        const v8f* acc = bb ? acc1 : acc0;
#pragma unroll
        for (int jt = 0; jt < 5; ++jt) {
            int jj = jt * 16 + ll;
#pragma unroll
            for (int r = 0; r < 8; ++r) {
                int n = n0 + r + lh * 8;
                float v = acc[jt][r];
                if (jj < 2) {
                    ob[n * OUTC + jj] = v + cls_b[jj];
                } else if (jj < JCOLS) {
                    int col = jj + 2;
                    ob[n * OUTC + col] = v + reg_b[jj - 2] + anchors[n * OUTC + col];
                }
            }
        }
        if (ll == 0) {      // anchor cols 2:4 passthrough
#pragma unroll
            for (int r = 0; r < 8; ++r) {
                int n = n0 + r + lh * 8;
                ob[n * OUTC + 2] = anchors[n * OUTC + 2];
                ob[n * OUTC + 3] = anchors[n * OUTC + 3];
            }
        }
    }
}

extern "C" void kernel_launch(void* const* d_in, const int* in_sizes, int n_in,
                              void* d_out, int out_size, void* d_ws, size_t ws_size,
                              hipStream_t stream) {
    const float*         x       = (const float*)d_in[0];
    const float*         conv_w  = (const float*)d_in[1];
    const float*         conv_b  = (const float*)d_in[2];
    const float*         cls_w   = (const float*)d_in[3];
    const float*         cls_b   = (const float*)d_in[4];
    const float*         reg_w   = (const float*)d_in[5];
    const float*         reg_b   = (const float*)d_in[6];
    const float*         anchors = (const float*)d_in[7];
    const int*           cut_xs  = (const int*)d_in[8];
    const unsigned char* invalid = (const unsigned char*)d_in[9];

    float* feat = (float*)d_ws;                       // 16*64*3600 f32 = 14.7 MB
    float* wp   = feat + (size_t)NB * CF * HW;        // 2880*80 f32 = 0.92 MB

    pack_w_kernel<<<(KDIM * JPAD + 255) / 256, 256, 0, stream>>>(cls_w, reg_w, wp);
    conv1x1_kernel<<<450, 256, 0, stream>>>(x, conv_w, conv_b, feat);   // 3600 waves
    head_gemm_kernel<<<174, 256, 0, stream>>>(feat, wp, cls_b, reg_b, anchors,
                                              cut_xs, invalid, (float*)d_out); // 1392 waves
}